// KerasModel_65180423684365
// MI455X (gfx1250) — compile-verified
//
#include <hip/hip_runtime.h>
#include <hip/hip_bf16.h>
#include <math.h>

typedef __attribute__((ext_vector_type(2))) float v2f;
typedef __attribute__((ext_vector_type(8))) float v8f;

#define N_FEAT   75
#define HID      128
#define BATCH    50
#define N_ATOMS  200000
#define BN_EPS   1e-3f
#define N_TILES  12502
#define NCHUNK   256
#define CHSZ     782            // 256*782 = 200192 >= 200000
#define KSTEPS   38             // 152 K in steps of 4 (self 76 | nbr 76)
#define WF_ELEMS (11 * KSTEPS * 2 * 128)   // 107008 fragment pairs
#define WF_FLOATS (WF_ELEMS * 2)           // 214016 floats (divisible by 256)

// Degree range starts (atoms are degree-sorted); last entry = N_ATOMS.
__device__ __constant__ int DEG_START_C[12] =
    {0, 1000, 20000, 60000, 120000, 170000, 185000, 193000, 197000, 199000, 199700, 200000};
// Cumulative 16-row tile counts per degree.
__device__ __constant__ int TILE_CUM_C[12] =
    {0, 63, 1251, 3751, 7501, 10626, 11564, 12064, 12314, 12439, 12483, 12502};

// ---------------------------------------------------------------------------
// Pre-swizzle weights into WMMA B-fragment order:
//   Wf[((d*38 + kstep)*2 + hs)*128 + col] = { W[widx][kb][col], W[widx][kb+1][col] }
// with kb = 4*kstep + 2*hs; zero-padded at K=75/151 and for d==0 nbr half.
// Makes the GEMM inner loop guard-free and degree-uniform.
// ---------------------------------------------------------------------------
__global__ __launch_bounds__(128)
void prep_wf_kernel(const float* __restrict__ W, float* __restrict__ Wf)
{
    const int idx = blockIdx.x * 128 + threadIdx.x;
    if (idx >= WF_ELEMS) return;
    const int col = idx & 127;
    int t = idx >> 7;
    const int hs = t & 1;  t >>= 1;
    const int kstep = t % KSTEPS;
    const int d = t / KSTEPS;

    const int kb = kstep * 4 + 2 * hs;
    const int isnbr = (kb >= 76);
    const int kw = isnbr ? (kb - 76) : kb;
    const int widx = (d == 0) ? 20 : (isnbr ? 2 * (d - 1) : 2 * (d - 1) + 1);

    float v0 = 0.f, v1 = 0.f;
    if (!(d == 0 && isnbr)) {
        const float* wp = W + widx * (N_FEAT * HID) + kw * HID + col;
        if (kw < 75)     v0 = wp[0];
        if (kw + 1 < 75) v1 = wp[HID];
    }
    Wf[(size_t)idx * 2]     = v0;
    Wf[(size_t)idx * 2 + 1] = v1;
}

// ---------------------------------------------------------------------------
// Fused: gather(nbr_sum) + per-degree dual GEMM + bias + tanh + batchnorm1
// One block = one 16-row tile of one degree range; 4 waves cover 128 out cols.
// ---------------------------------------------------------------------------
__global__ __launch_bounds__(128)
void gc1_fused_kernel(const float* __restrict__ feat,     // [N_ATOMS, 75]
                      const float* __restrict__ Wf,       // pre-swizzled fragments
                      const float* __restrict__ Bv,       // [21, 128]
                      const float* __restrict__ bg, const float* __restrict__ bb,
                      const float* __restrict__ bm, const float* __restrict__ bvr,
                      const int* __restrict__ adj1,  const int* __restrict__ adj2,
                      const int* __restrict__ adj3,  const int* __restrict__ adj4,
                      const int* __restrict__ adj5,  const int* __restrict__ adj6,
                      const int* __restrict__ adj7,  const int* __restrict__ adj8,
                      const int* __restrict__ adj9,  const int* __restrict__ adj10,
                      float* __restrict__ bn1)              // [N_ATOMS, 128]
{
    __shared__ float As[16 * 156];   // 16 rows x 152 K (stride 156: bank-conflict free)
    __shared__ int   adjL[160];      // up to 16 rows x 10 neighbor indices

    const int tile = blockIdx.x;
    int d = 0;
#pragma unroll
    for (int i = 1; i < 11; i++)
        if (tile >= TILE_CUM_C[i]) d = i;
    const int row0 = DEG_START_C[d] + (tile - TILE_CUM_C[d]) * 16;
    int rows = DEG_START_C[d + 1] - row0;
    if (rows > 16) rows = 16;

    const int* adj = nullptr;
    switch (d) {
        case 1: adj = adj1; break;  case 2: adj = adj2; break;
        case 3: adj = adj3; break;  case 4: adj = adj4; break;
        case 5: adj = adj5; break;  case 6: adj = adj6; break;
        case 7: adj = adj7; break;  case 8: adj = adj8; break;
        case 9: adj = adj9; break;  case 10: adj = adj10; break;
        default: break;
    }

    const int tid = threadIdx.x;

    // Stage adjacency rows for this tile into LDS (reused ~75x per row).
    if (d > 0) {
        const int na = rows * d;
        const int base = (row0 - DEG_START_C[d]) * d;
        for (int i = tid; i < na; i += 128) adjL[i] = adj[base + i];
    }
    __syncthreads();

    // Stage A panel: cols [0,75)=self feats, [76,151)=neighbor sums; pads zeroed.
    for (int i = tid; i < 16 * 152; i += 128) {
        const int m = i / 152;
        const int k = i - m * 152;
        float v = 0.f;
        if (m < rows) {
            if (k < 75) {
                v = feat[(row0 + m) * N_FEAT + k];
            } else if (k >= 76 && k < 151 && d > 0) {
                const int kk = k - 76;
                const int* ap = adjL + m * d;
                float s = 0.f;
                for (int j = 0; j < d; j++) s += feat[ap[j] * N_FEAT + kk];
                v = s;
            }
        }
        As[m * 156 + k] = v;
    }
    __syncthreads();

    const int wave = tid >> 5, lane = tid & 31;
    const int hs = lane >> 4, l16 = lane & 15;
    const int nb = wave * 32;
    const int col0 = nb + l16, col1 = col0 + 16;

    // Guard-free, degree-uniform K loop over pre-swizzled fragments.
    const float* wfp = Wf + ((size_t)((d * KSTEPS) * 2 + hs) * 128 + col0) * 2;
    const float* ap  = &As[l16 * 156 + (hs << 1)];

    v8f c0 = {}, c1 = {};
#pragma unroll 4
    for (int ks = 0; ks < KSTEPS; ks++) {
        const v2f a   = *(const v2f*)(ap + ks * 4);
        const v2f bb0 = *(const v2f*)(wfp);
        const v2f bb1 = *(const v2f*)(wfp + 32);
        wfp += 512;
        c0 = __builtin_amdgcn_wmma_f32_16x16x4_f32(false, a, false, bb0, (short)0, c0, false, false);
        c1 = __builtin_amdgcn_wmma_f32_16x16x4_f32(false, a, false, bb1, (short)0, c1, false, false);
    }

    const int wself = (d == 0) ? 20 : (2 * (d - 1) + 1);
    const int wrel  = (d == 0) ? 20 : (2 * (d - 1));
    float bias0, bias1;
    if (d == 0) {
        bias0 = Bv[20 * HID + col0];
        bias1 = Bv[20 * HID + col1];
    } else {
        bias0 = Bv[wrel * HID + col0] + Bv[wself * HID + col0];
        bias1 = Bv[wrel * HID + col1] + Bv[wself * HID + col1];
    }
    const float sc0 = bg[col0] * rsqrtf(bvr[col0] + BN_EPS);
    const float sh0 = bb[col0] - bm[col0] * sc0;
    const float sc1 = bg[col1] * rsqrtf(bvr[col1] + BN_EPS);
    const float sh1 = bb[col1] - bm[col1] * sc1;

#pragma unroll
    for (int r = 0; r < 8; r++) {
        const int m = r + hs * 8;         // C/D layout: M = r + 8*(lane>=16), N = lane%16
        if (m < rows) {
            const int row = row0 + m;
            bn1[row * HID + col0] = tanhf(c0[r] + bias0) * sc0 + sh0;
            bn1[row * HID + col1] = tanhf(c1[r] + bias1) * sc1 + sh1;
        }
    }
}

// ---------------------------------------------------------------------------
// Deterministic segment sum/max, phase 1: per-chunk partials in private LDS
// columns (thread t exclusively owns column t -> no races, no barriers).
// ---------------------------------------------------------------------------
__global__ __launch_bounds__(128)
void seg_partial_kernel(const float* __restrict__ bn1, const int* __restrict__ mem,
                        float* __restrict__ psum, float* __restrict__ pmax)
{
    __shared__ float ls[BATCH * HID];
    __shared__ float lm[BATCH * HID];
    const int t = threadIdx.x;
    for (int s = t; s < BATCH * HID; s += 128) { ls[s] = 0.f; lm[s] = -INFINITY; }

    const int c = blockIdx.x;
    const int a0 = c * CHSZ;
    int a1 = a0 + CHSZ;
    if (a1 > N_ATOMS) a1 = N_ATOMS;

    for (int a = a0; a < a1; a++) {
        const int m = mem[a];
        const float v = bn1[a * HID + t];
        ls[m * HID + t] += v;
        lm[m * HID + t] = fmaxf(lm[m * HID + t], v);
    }
    for (int s = 0; s < BATCH; s++) {
        psum[(c * BATCH + s) * HID + t] = ls[s * HID + t];
        pmax[(c * BATCH + s) * HID + t] = lm[s * HID + t];
    }
}

// Phase 2: fixed-order reduction over chunks + tanh -> readout [50, 256]
__global__ __launch_bounds__(128)
void seg_reduce_kernel(const float* __restrict__ psum, const float* __restrict__ pmax,
                       float* __restrict__ readout)
{
    const int s = blockIdx.x;
    const int t = threadIdx.x;
    float sum = 0.f, mx = -INFINITY;
    for (int c = 0; c < NCHUNK; c++) {
        sum += psum[(c * BATCH + s) * HID + t];
        mx = fmaxf(mx, pmax[(c * BATCH + s) * HID + t]);
    }
    readout[s * 256 + t]       = tanhf(sum);
    readout[s * 256 + 128 + t] = tanhf(mx);
}

// Head: logits = readout @ d2_W + d2_b, pairwise softmax over last dim of [12,2]
__global__ __launch_bounds__(32)
void head_kernel(const float* __restrict__ readout, const float* __restrict__ W2,
                 const float* __restrict__ b2, float* __restrict__ out)
{
    const int s = blockIdx.x;
    const int lane = threadIdx.x;
    float dot = 0.f;
    if (lane < 24) {
        dot = b2[lane];
        for (int k = 0; k < 256; k++) dot += readout[s * 256 + k] * W2[k * 24 + lane];
    }
    const float other = __shfl_xor(dot, 1, 32);
    if (lane < 24) {
        const float m = fmaxf(dot, other);
        const float e = expf(dot - m);
        const float eo = expf(other - m);
        out[s * 24 + lane] = e / (e + eo);
    }
}

extern "C" void kernel_launch(void* const* d_in, const int* in_sizes, int n_in,
                              void* d_out, int out_size, void* d_ws, size_t ws_size,
                              hipStream_t stream)
{
    (void)in_sizes; (void)n_in; (void)out_size; (void)ws_size;

    const float* feat       = (const float*)d_in[0];
    const int*   membership = (const int*)d_in[2];
    const int*   adj[10];
    for (int i = 0; i < 10; i++) adj[i] = (const int*)d_in[3 + i];
    const float* gc1_W = (const float*)d_in[13];
    const float* gc1_b = (const float*)d_in[14];
    const float* bn1_g = (const float*)d_in[17];
    const float* bn1_b = (const float*)d_in[18];
    const float* bn1_m = (const float*)d_in[19];
    const float* bn1_v = (const float*)d_in[20];
    const float* d2_W  = (const float*)d_in[31];
    const float* d2_b  = (const float*)d_in[32];

    float* ws      = (float*)d_ws;
    float* Wf      = ws;                                   // 214016 floats (856KB)
    float* bn1     = Wf + WF_FLOATS;                       // 200000*128
    float* psum    = bn1 + (size_t)N_ATOMS * HID;          // 256*50*128
    float* pmax    = psum + (size_t)NCHUNK * BATCH * HID;  // 256*50*128
    float* readout = pmax + (size_t)NCHUNK * BATCH * HID;  // 50*256

    prep_wf_kernel<<<(WF_ELEMS + 127) / 128, 128, 0, stream>>>(gc1_W, Wf);

    gc1_fused_kernel<<<N_TILES, 128, 0, stream>>>(
        feat, Wf, gc1_b, bn1_g, bn1_b, bn1_m, bn1_v,
        adj[0], adj[1], adj[2], adj[3], adj[4], adj[5], adj[6], adj[7], adj[8], adj[9],
        bn1);

    seg_partial_kernel<<<NCHUNK, 128, 0, stream>>>(bn1, membership, psum, pmax);
    seg_reduce_kernel<<<BATCH, 128, 0, stream>>>(psum, pmax, readout);
    head_kernel<<<BATCH, 32, 0, stream>>>(readout, d2_W, d2_b, (float*)d_out);
}